// InformationGainObjectDetection_23785528885845
// MI455X (gfx1250) — compile-verified
//
#include <hip/hip_runtime.h>
#include <hip/hip_bf16.h>

// ---------------------------------------------------------------------------
// InformationGainObjectDetection for MI455X (gfx1250)
//   H=W=2048, N_BOX=M_BOX=512, SUBSAMPLE=2 -> mask 1024x1024, out 2048x2048
// Pipeline:
//   K0 zero      : clear mask + matched_prev flags
//   K1 match     : 512x512 IoU + first-argmax (boxes_prev staged to LDS via
//                  async global->LDS), emits boxes_all[0..1024) / vals_all
//   K2 scatter   : 1536 rect fills via GLOBAL_ATOMIC_MAX_U32 on f32 bits
//   K3 upsample  : nearest 2x via V_WMMA_F32_16X16X4_F32 (exact 0/1 matmul)
// ---------------------------------------------------------------------------

#define H2 1024
#define W2 1024

typedef float v2f __attribute__((ext_vector_type(2)));
typedef float v8f __attribute__((ext_vector_type(8)));
typedef int   v4i __attribute__((vector_size(4 * sizeof(int))));  // matches builtin param

#define AS1 __attribute__((address_space(1)))
#define AS3 __attribute__((address_space(3)))

#if defined(__HIP_DEVICE_COMPILE__)
# if __has_builtin(__builtin_amdgcn_wmma_f32_16x16x4_f32)
#  define HAVE_WMMA_F32 1
# else
#  define HAVE_WMMA_F32 0
#  warning "gfx1250: __builtin_amdgcn_wmma_f32_16x16x4_f32 NOT available; scalar upsample fallback"
# endif
# if __has_builtin(__builtin_amdgcn_global_load_async_to_lds_b128)
#  define HAVE_ASYNC_LDS 1
# else
#  define HAVE_ASYNC_LDS 0
#  warning "gfx1250: __builtin_amdgcn_global_load_async_to_lds_b128 NOT available; trying inline asm"
# endif
#else
# define HAVE_WMMA_F32 0
# define HAVE_ASYNC_LDS 0
#endif

// ------------------------------- K0: zero ----------------------------------
__global__ void ig_zero_kernel(unsigned* __restrict__ p, int n) {
  int i = blockIdx.x * blockDim.x + threadIdx.x;
  if (i < n) p[i] = 0u;
}

// ------------------------------- K1: match ---------------------------------
// One block of 512 threads: thread i handles current box i.
__global__ __launch_bounds__(512) void ig_match_kernel(
    const int* __restrict__ boxes_cur, const float* __restrict__ scores_cur,
    const int* __restrict__ boxes_prev, const float* __restrict__ scores_prev,
    int4* __restrict__ boxes_all, float* __restrict__ vals_all,
    int* __restrict__ flags) {
  __shared__ int4   s_raw[512];   // raw boxes_prev rows
  __shared__ float4 s_b[512];     // subsampled prev boxes, as float
  __shared__ float  s_area[512];  // prev areas

  const int i = threadIdx.x;

  // ---- stage boxes_prev (8 KB) into LDS via async global->LDS ----
#if HAVE_ASYNC_LDS
  __builtin_amdgcn_global_load_async_to_lds_b128(
      (AS1 v4i*)(boxes_prev + 4 * i),
      (AS3 v4i*)&s_raw[i],
      /*offset=*/0, /*cpol=*/0);
# if __has_builtin(__builtin_amdgcn_s_wait_asynccnt)
  __builtin_amdgcn_s_wait_asynccnt(0);
# else
  asm volatile("s_wait_asynccnt 0" ::: "memory");
# endif
#elif defined(__HIP_DEVICE_COMPILE__)
  {
    unsigned lds_off = (unsigned)(size_t)&s_raw[i];
    const int* g = boxes_prev + 4 * i;
    asm volatile("global_load_async_to_lds_b128 %0, %1, off"
                 :: "v"(lds_off), "v"(g) : "memory");
    asm volatile("s_wait_asynccnt 0" ::: "memory");
  }
#else
  s_raw[i] = ((const int4*)boxes_prev)[i];
#endif

  // subsample (x>>1 == floor-div-2 for nonneg) and precompute areas
  {
    int4 r = s_raw[i];
    float bx1 = (float)(r.x >> 1), by1 = (float)(r.y >> 1);
    float bx2 = (float)(r.z >> 1), by2 = (float)(r.w >> 1);
    s_b[i] = make_float4(bx1, by1, bx2, by2);
    s_area[i] = (bx2 - bx1) * (by2 - by1);
  }
  __syncthreads();

  int4 rc = ((const int4*)boxes_cur)[i];
  int4 bc = make_int4(rc.x >> 1, rc.y >> 1, rc.z >> 1, rc.w >> 1);
  const float ax1 = (float)bc.x, ay1 = (float)bc.y;
  const float ax2 = (float)bc.z, ay2 = (float)bc.w;
  const float area_a = (ax2 - ax1) * (ay2 - ay1);

  float best = 0.0f;
  int   jbest = 0;
  bool  matched = false;
  for (int j = 0; j < 512; ++j) {          // strict '>' keeps FIRST max, like argmax
    float4 bb = s_b[j];
    float xl = fmaxf(ax1, bb.x), yt = fmaxf(ay1, bb.y);
    float xr = fminf(ax2, bb.z), yb = fminf(ay2, bb.w);
    float inter = fmaxf(xr - xl, 0.0f) * fmaxf(yb - yt, 0.0f);
    float iou = inter / (area_a + s_area[j] - inter);
    if (iou > best) { best = iou; jbest = j; matched = true; }
  }

  const float ig = 1.0f - best;
  boxes_all[i] = bc;
  vals_all[i]  = ig * scores_cur[i];
  if (matched) {
    float4 bb = s_b[jbest];
    boxes_all[512 + i] = make_int4((int)bb.x, (int)bb.y, (int)bb.z, (int)bb.w);
    vals_all[512 + i]  = ig * scores_prev[jbest];
    atomicExch(&flags[jbest], 1);          // mark prev box as matched
  } else {
    boxes_all[512 + i] = make_int4(0, 0, 0, 0);
    vals_all[512 + i]  = 0.0f;
  }
}

// ------------------------------ K2: scatter --------------------------------
// One block per rectangle. Values are >= 0, so f32 max == u32 max on bits.
__global__ __launch_bounds__(256) void ig_scatter_kernel(
    const int4* __restrict__ boxes_all, const float* __restrict__ vals_all,
    const int* __restrict__ boxes_prev, const float* __restrict__ scores_prev,
    const int* __restrict__ flags, unsigned* __restrict__ mask, int scale) {
  const int b = blockIdx.x;
  int4 box; float val;
  if (b < 1024) {
    box = boxes_all[b];
    val = vals_all[b];
  } else {
    const int j = b - 1024;                 // write 3: unmatched prev boxes
    int4 r = ((const int4*)boxes_prev)[j];
    box = make_int4(r.x >> 1, r.y >> 1, r.z >> 1, r.w >> 1);
    val = flags[j] ? 0.0f : scores_prev[j];
  }
  if (!(val > 0.0f)) return;                // val==0 is a no-op under max
  const unsigned bits = __float_as_uint(val);
  const int x1 = box.x * scale, y1 = box.y * scale;
  const int x2 = box.z * scale, y2 = box.w * scale;
  const int W = W2 * scale;
  for (int y = y1; y < y2; ++y) {
    unsigned* rowp = mask + (size_t)y * W;
    for (int x = x1 + (int)threadIdx.x; x < x2; x += 256)
      atomicMax(rowp + x, bits);            // -> GLOBAL_ATOMIC_MAX_U32
  }
}

// ------------------------- K3: upsample via WMMA ---------------------------
// out(2048x2048) = nearest-2x(mask(1024x1024)).
// Per 16x16 mask tile: D_left = A0*QA + A1*QB, D_right = A2*QA + A3*QB,
// where QA[k][c]=( (c>>1)==k ), QB[k][c]=( (c>>1)==k+4 ) are exact 0/1 f32
// matrices -> column duplication is exact. Row duplication happens at store.
__global__ __launch_bounds__(256) void ig_upsample_kernel(
    const float* __restrict__ mask, float* __restrict__ out) {
#if HAVE_WMMA_F32
  const int lane = threadIdx.x & 31;
  const int wave = threadIdx.x >> 5;
  const int tile = blockIdx.x * 8 + wave;   // 4096 tiles = 512 blocks * 8 waves
  const int ty = tile >> 6, tx = tile & 63;
  const int row = lane & 15;
  const int kh  = lane >> 4;                // K half: 0 or 1
  const float* base = mask + ((size_t)(ty * 16 + row)) * W2 + tx * 16;

  // A_t (16x4 f32): lane holds in[row][4t + 2*kh + {0,1}]  (one b64 load each)
  v2f a0 = *(const v2f*)(base + 0 * 4 + 2 * kh);
  v2f a1 = *(const v2f*)(base + 1 * 4 + 2 * kh);
  v2f a2 = *(const v2f*)(base + 2 * 4 + 2 * kh);
  v2f a3 = *(const v2f*)(base + 3 * 4 + 2 * kh);

  // B (4x16 f32): lanes 0-15 hold K=0,1 ; lanes 16-31 hold K=2,3
  const int col = lane & 15;
  const int c2  = col >> 1;
  v2f bA, bB;
  bA.x = (c2 == (2 * kh + 0)) ? 1.0f : 0.0f;
  bA.y = (c2 == (2 * kh + 1)) ? 1.0f : 0.0f;
  bB.x = (c2 == (2 * kh + 4)) ? 1.0f : 0.0f;
  bB.y = (c2 == (2 * kh + 5)) ? 1.0f : 0.0f;

  v8f z = {0.f, 0.f, 0.f, 0.f, 0.f, 0.f, 0.f, 0.f};
  v8f dL = __builtin_amdgcn_wmma_f32_16x16x4_f32(false, a0, false, bA, (short)0, z,  false, false);
  dL     = __builtin_amdgcn_wmma_f32_16x16x4_f32(false, a1, false, bB, (short)0, dL, false, false);
  v8f dR = __builtin_amdgcn_wmma_f32_16x16x4_f32(false, a2, false, bA, (short)0, z,  false, false);
  dR     = __builtin_amdgcn_wmma_f32_16x16x4_f32(false, a3, false, bB, (short)0, dR, false, false);

  // D layout: VGPR r, lanes 0-15 -> (M=r, N=lane); lanes 16-31 -> (M=r+8, N=lane-16)
  float* obase = out + ((size_t)(ty * 32)) * 2048 + tx * 32;
#pragma unroll
  for (int r = 0; r < 8; ++r) {
    const int M = r + 8 * kh;
    const float vL = dL[r], vR = dR[r];
    float* p0 = obase + (size_t)(2 * M) * 2048 + col;
    float* p1 = p0 + 2048;
    p0[0] = vL; p0[16] = vR;                // row 2M
    p1[0] = vL; p1[16] = vR;                // row 2M+1 (nearest dup)
  }
#else
  // scalar fallback (also what the host pass parses)
  const int tid = blockIdx.x * blockDim.x + threadIdx.x;  // 131072 threads
  for (int p = tid; p < H2 * W2; p += 512 * 256) {
    const int y = p >> 10, x = p & (W2 - 1);
    const float v = mask[p];
    float* o = out + (size_t)(2 * y) * 2048 + 2 * x;
    v2f vv = {v, v};
    *(v2f*)o = vv;
    *(v2f*)(o + 2048) = vv;
  }
#endif
}

// ------------------------------- launcher ----------------------------------
extern "C" void kernel_launch(void* const* d_in, const int* in_sizes, int n_in,
                              void* d_out, int out_size, void* d_ws, size_t ws_size,
                              hipStream_t stream) {
  (void)in_sizes; (void)n_in; (void)out_size;
  const int*   boxes_cur   = (const int*)d_in[1];
  const float* scores_cur  = (const float*)d_in[2];
  const int*   boxes_prev  = (const int*)d_in[3];
  const float* scores_prev = (const float*)d_in[4];
  float* out = (float*)d_out;
  char*  ws  = (char*)d_ws;

  const size_t MASK_BYTES = (size_t)H2 * W2 * sizeof(float);
  const bool half = ws_size >= MASK_BYTES + 32 * 1024;  // half-res path needs 4MB+meta

  float* mask;  size_t metaOff;
  int scale, maskElems;
  if (half) { mask = (float*)ws; metaOff = MASK_BYTES; scale = 1; maskElems = H2 * W2; }
  else      { mask = out;        metaOff = 0;          scale = 2; maskElems = 2048 * 2048; }

  int4*  boxes_all = (int4*)(ws + metaOff);
  float* vals_all  = (float*)(ws + metaOff + 1024 * sizeof(int4));
  int*   flags     = (int*)(ws + metaOff + 1024 * sizeof(int4) + 1024 * sizeof(float));

  ig_zero_kernel<<<(maskElems + 511) / 512, 512, 0, stream>>>((unsigned*)mask, maskElems);
  ig_zero_kernel<<<1, 512, 0, stream>>>((unsigned*)flags, 512);

  ig_match_kernel<<<1, 512, 0, stream>>>(boxes_cur, scores_cur, boxes_prev, scores_prev,
                                         boxes_all, vals_all, flags);

  ig_scatter_kernel<<<1536, 256, 0, stream>>>(boxes_all, vals_all, boxes_prev, scores_prev,
                                              flags, (unsigned*)mask, scale);

  if (half)
    ig_upsample_kernel<<<512, 256, 0, stream>>>(mask, out);
}